// ROIRelationHead_34668976014004
// MI455X (gfx1250) — compile-verified
//
#include <hip/hip_runtime.h>
#include <math.h>

// ---------------------------------------------------------------------------
// Problem constants (from reference)
// ---------------------------------------------------------------------------
#define N_INST   2048
#define N_REL    1024
#define KTOT     50176              // C*H*W = 256*14*14
#define KSLICES  32                 // K split across waves
#define KSLAB    (KTOT / KSLICES)   // 1568 floats per slab
#define TILES    (N_INST / 16)      // 128 instance tiles of 16

typedef __attribute__((ext_vector_type(2))) float v2f;
typedef __attribute__((ext_vector_type(8))) float v8f;

// ---------------------------------------------------------------------------
// Phase 1: streaming GEMV via V_WMMA_F32_16X16X4_F32.
//   A (16x4)  : row 0 = w_lo[k0..k0+3], row 1 = w_hi[k0..k0+3], rows 2..15 = 0
//   B (4x16)  : B(k,n) = feat[tile*16+n][k0+k]
//   D (16x16) : D(0,n) = partial pA[n], D(1,n) = partial pB[n]
// A-operand lane layout (ISA 7.12.2, 32-bit 16x4): lane (khalf*16 + m) holds
// K = 2*khalf + vgpr  -> one b64 load per lane per WMMA. B mirrored.
// Feature loads are NON-TEMPORAL: 411 MB read exactly once; w (400 KB) is
// regular-temporal so its ~51 MB of logical reuse is served from L2.
// One wave per block (EXEC all-ones, required by WMMA). grid = 128 tiles x 32
// k-slices = 4096 waves; each wave does KSLAB/8 = 196 iterations of 2 WMMAs
// with independent accumulators to hide XDL latency.
// ---------------------------------------------------------------------------
__global__ __launch_bounds__(32)
void rel_gemv_wmma(const float* __restrict__ feat,
                   const float* __restrict__ w,
                   float2* __restrict__ part)
{
    const int wave  = blockIdx.x;
    const int tile  = wave / KSLICES;   // which 16 instances
    const int ks    = wave % KSLICES;   // which K slab
    const int lane  = threadIdx.x;      // 0..31
    const int nloc  = lane & 15;        // M (for A) / N (for B)
    const int khalf = lane >> 4;        // 0: K={0,1}, 1: K={2,3}

    // B: feature stream. Row stride 50176 floats = 200704 B (128B-aligned).
    const float* fptr = feat + (size_t)(tile * 16 + nloc) * KTOT
                             + (size_t)ks * KSLAB + 2 * khalf;
    // A: weight rows. Row 0 -> w_lo, row 1 -> w_hi, rows >=2 masked to zero.
    const float* wptr = w + (nloc == 1 ? KTOT : 0)
                          + (size_t)ks * KSLAB + 2 * khalf;
    const bool wvalid = (nloc < 2);

    v8f c0 = {0.f,0.f,0.f,0.f,0.f,0.f,0.f,0.f};
    v8f c1 = {0.f,0.f,0.f,0.f,0.f,0.f,0.f,0.f};

    #pragma unroll 2
    for (int it = 0; it < KSLAB / 8; ++it) {
        // Single-use stream -> non-temporal hint on the global_load_b64s.
        v2f b0 = __builtin_nontemporal_load((const v2f*)(fptr));     // K k0..k0+3
        v2f b1 = __builtin_nontemporal_load((const v2f*)(fptr + 4)); // K k0+4..k0+7
        v2f a0 = {0.f, 0.f};
        v2f a1 = {0.f, 0.f};
        if (wvalid) {                   // exec-masked loads; others stay 0
            a0 = *(const v2f*)(wptr);
            a1 = *(const v2f*)(wptr + 4);
        }
        // 8 args: (neg_a, A, neg_b, B, c_mod, C, reuse_a, reuse_b)
        c0 = __builtin_amdgcn_wmma_f32_16x16x4_f32(false, a0, false, b0,
                                                   (short)0, c0, false, false);
        c1 = __builtin_amdgcn_wmma_f32_16x16x4_f32(false, a1, false, b1,
                                                   (short)0, c1, false, false);
        fptr += 8;
        wptr += 8;
    }

    // C/D layout: VGPR v holds M=v on lanes 0-15 (N=lane). We need M=0 (pA)
    // and M=1 (pB) -> c[0], c[1] on lanes 0-15. 16 lanes store float2 each:
    // one fully coalesced 128B store per wave (RT: phase 2 re-reads via L2).
    if (lane < 16) {
        float2 r;
        r.x = c0[0] + c1[0];   // pA partial
        r.y = c0[1] + c1[1];   // pB partial
        part[(size_t)ks * N_INST + tile * 16 + lane] = r;
    }
}

// ---------------------------------------------------------------------------
// Phase 2: reduce 32 k-slab partials per instance, then BCE over 1024 pairs.
// Single 1024-thread block; fixed-order tree reduction => deterministic.
//   target=1 sample: -log_sigmoid(l1)  = softplus(-l1)
//   target=0 sample: -log_sigmoid(-l2) = softplus( l2)
// ---------------------------------------------------------------------------
__device__ __forceinline__ float softplus_stable(float x)
{
    float ax = fabsf(x);
    return fmaxf(x, 0.f) + log1pf(__expf(-ax));
}

__global__ __launch_bounds__(1024)
void rel_loss(const float2* __restrict__ part,
              const int*    __restrict__ pairs,
              const float*  __restrict__ bias,
              float*        __restrict__ out)
{
    __shared__ float2 P[N_INST];      // 16 KB
    __shared__ float  red[1024];      // 4 KB

    const int tid = threadIdx.x;

    // Reduce partials: each thread sums 2 instances over 32 slabs.
    for (int n = tid; n < N_INST; n += 1024) {
        float sx = 0.f, sy = 0.f;
        #pragma unroll 4
        for (int ks = 0; ks < KSLICES; ++ks) {
            float2 v = part[(size_t)ks * N_INST + n];
            sx += v.x;
            sy += v.y;
        }
        float2 acc; acc.x = sx; acc.y = sy;
        P[n] = acc;
    }
    __syncthreads();

    // One pair per thread.
    const int   i = pairs[2 * tid];
    const int   j = pairs[2 * tid + 1];
    const float b = bias[0];
    const float l1 = P[i].x + P[j].y + b;   // (i,j) sample, target 1
    const float l2 = P[j].x + P[i].y + b;   // (j,i) sample, target 0
    red[tid] = softplus_stable(-l1) + softplus_stable(l2);
    __syncthreads();

    // Deterministic tree reduction over 1024 lanes.
    for (int s = 512; s > 0; s >>= 1) {
        if (tid < s) red[tid] += red[tid + s];
        __syncthreads();
    }
    if (tid == 0) out[0] = red[0] * (1.0f / (2.0f * N_REL));
}

// ---------------------------------------------------------------------------
// Entry point. Inputs (setup_inputs order): mask_features f32, pairs int,
// w f32, b f32. Workspace: 32*2048 float2 partials = 512 KB, fully written
// by phase 1 before phase 2 reads it (no init / no cross-call state).
// ---------------------------------------------------------------------------
extern "C" void kernel_launch(void* const* d_in, const int* in_sizes, int n_in,
                              void* d_out, int out_size, void* d_ws, size_t ws_size,
                              hipStream_t stream)
{
    const float* feat  = (const float*)d_in[0];
    const int*   pairs = (const int*)  d_in[1];
    const float* w     = (const float*)d_in[2];
    const float* b     = (const float*)d_in[3];

    float2* part = (float2*)d_ws;

    rel_gemv_wmma<<<TILES * KSLICES, 32, 0, stream>>>(feat, w, part);
    rel_loss<<<1, 1024, 0, stream>>>(part, pairs, b, (float*)d_out);
}